// SimplePoseGNN_4183298146474
// MI455X (gfx1250) — compile-verified
//
#include <hip/hip_runtime.h>

#define HN 256            // hidden size H
#define EPSF 1e-5f

typedef __attribute__((ext_vector_type(2))) float v2f;
typedef __attribute__((ext_vector_type(8))) float v8f;

// ---------------------------------------------------------------- utilities
__global__ void zero_f32(float* __restrict__ p, long n) {
  long i = (long)blockIdx.x * blockDim.x + threadIdx.x;
  long stride = (long)gridDim.x * blockDim.x;
  for (; i < n; i += stride) p[i] = 0.0f;
}

__global__ void degree_kernel(const int* __restrict__ src, const int* __restrict__ dst,
                              float* __restrict__ deg_out, float* __restrict__ deg_in, int E) {
  int e = blockIdx.x * blockDim.x + threadIdx.x;
  if (e >= E) return;
  atomicAdd(&deg_out[src[e]], 1.0f);
  atomicAdd(&deg_in[dst[e]], 1.0f);
}

__global__ void rsqrt_clip_kernel(float* __restrict__ p, int n) {
  int i = blockIdx.x * blockDim.x + threadIdx.x;
  if (i >= n) return;
  p[i] = rsqrtf(fmaxf(p[i], 1.0f));
}

// x[n,h] = nf[n,0]*W_emb[0,h] + nf[n,1]*W_emb[1,h] + b_emb[h]
__global__ void embed_kernel(const float* __restrict__ nf, const float* __restrict__ W_emb,
                             const float* __restrict__ b_emb, float* __restrict__ x, int N) {
  long t = (long)blockIdx.x * blockDim.x + threadIdx.x;
  if (t >= (long)N * HN) return;
  int n = (int)(t >> 8);
  int h = (int)(t & (HN - 1));
  x[t] = nf[n * 2] * W_emb[h] + nf[n * 2 + 1] * W_emb[HN + h] + b_emb[h];
}

// agg[dst] += x[src] * (n_out[src] * n_in[dst])
// (n_in folded into the edge scale by linearity, so the GEMM's A needs no scaling)
__global__ void scatter_kernel(const float* __restrict__ x, const float* __restrict__ nrm_out,
                               const float* __restrict__ nrm_in,
                               const int* __restrict__ src, const int* __restrict__ dst,
                               float* __restrict__ agg, int E) {
  long t = (long)blockIdx.x * blockDim.x + threadIdx.x;
  if (t >= (long)E * (HN / 4)) return;
  int e  = (int)(t >> 6);             // H/4 = 64 chunks per edge
  int h4 = (int)(t & 63) << 2;
  int s = src[e], d = dst[e];
  float sc = nrm_out[s] * nrm_in[d];
  const float4 xv = *(const float4*)(x + (size_t)s * HN + h4);
  float* a = agg + (size_t)d * HN + h4;
  atomicAdd(a + 0, xv.x * sc);
  atomicAdd(a + 1, xv.y * sc);
  atomicAdd(a + 2, xv.z * sc);
  atomicAdd(a + 3, xv.w * sc);
}

// ------------------------------------------------------- fused WMMA f32 GEMM
// out = epilogue( A @ B + bias ), K fixed at 256, B row-major [K, ncols].
// Epilogue: bias -> [BN(i0); relu] -> [BN(i1); relu] -> (+ residual)
// One wave computes a 16 x (16*NT) tile: one A fragment feeds NT chained
// V_WMMA_F32_16X16X4_F32 per k-step (NT accumulators live in VGPRs).
template <int NT, int NBN, bool HAS_RES, bool MASK_N>
__global__ __launch_bounds__(256) void wmma_gemm_fused(
    const float* __restrict__ A, const float* __restrict__ B, const float* __restrict__ bias,
    const float* __restrict__ bnG, const float* __restrict__ bnB,
    const float* __restrict__ bnM, const float* __restrict__ bnV,
    int bi0, int bi1,
    const float* __restrict__ res, float* __restrict__ out,
    int rows, int ncols) {
  const int lane = threadIdx.x & 31;
  const int wid  = (int)(((long)blockIdx.x * blockDim.x + threadIdx.x) >> 5);
  const int tiles_n = (ncols + 15) >> 4;
  const int gtn = (tiles_n + NT - 1) / NT;     // groups of NT column tiles
  const int tiles_m = rows >> 4;
  if (wid >= tiles_m * gtn) return;            // wave-uniform: EXEC stays full
  const int tm = wid / gtn;
  const int tg = wid - tm * gtn;
  const int l16   = lane & 15;
  const int khalf = (lane >> 4) << 1;          // 0 or 2 (A/B K sub-lane offset)
  const int arow  = tm * 16 + l16;             // A: M = lane%16

  const float* __restrict__ Ar = A + (size_t)arow * HN + khalf;

  int  colb[NT];
  bool cok[NT];
#pragma unroll
  for (int t = 0; t < NT; ++t) {
    colb[t] = (tg * NT + t) * 16 + l16;        // B: N = lane%16
    cok[t]  = (!MASK_N) || (colb[t] < ncols);
  }

  const v8f vzero = {};
  v8f acc[NT];
#pragma unroll
  for (int t = 0; t < NT; ++t) acc[t] = vzero;

  for (int kk = 0; kk < HN; kk += 4) {
    v2f a;
    a[0] = Ar[kk + 0];
    a[1] = Ar[kk + 1];
    const float* __restrict__ Bk = B + (size_t)(kk + khalf) * ncols;
#pragma unroll
    for (int t = 0; t < NT; ++t) {
      v2f b;
      if (cok[t]) {
        b[0] = Bk[colb[t]];
        b[1] = Bk[(size_t)ncols + colb[t]];
      } else {
        b[0] = 0.0f; b[1] = 0.0f;
      }
      acc[t] = __builtin_amdgcn_wmma_f32_16x16x4_f32(
          /*neg_a=*/false, a, /*neg_b=*/false, b,
          /*c_mod=*/(short)0, acc[t], /*reuse_a=*/false, /*reuse_b=*/false);
    }
  }

  // epilogue: per-lane column-constant params per tile
  const int mhalf = (lane >> 4) << 3;          // C/D: M = v + 8*(lane>=16)
#pragma unroll
  for (int t = 0; t < NT; ++t) {
    const int c = colb[t];
    float bia = 0.0f;
    float g0 = 1.0f, be0 = 0.0f, m0 = 0.0f, iv0 = 1.0f;
    float g1 = 1.0f, be1 = 0.0f, m1 = 0.0f, iv1 = 1.0f;
    if (cok[t]) {
      bia = bias[c];
      if constexpr (NBN >= 1) {
        g0 = bnG[bi0 * HN + c]; be0 = bnB[bi0 * HN + c];
        m0 = bnM[bi0 * HN + c]; iv0 = rsqrtf(bnV[bi0 * HN + c] + EPSF);
      }
      if constexpr (NBN >= 2) {
        g1 = bnG[bi1 * HN + c]; be1 = bnB[bi1 * HN + c];
        m1 = bnM[bi1 * HN + c]; iv1 = rsqrtf(bnV[bi1 * HN + c] + EPSF);
      }
    }
#pragma unroll
    for (int v = 0; v < 8; ++v) {
      const int r = tm * 16 + mhalf + v;
      float y = acc[t][v] + bia;
      if constexpr (NBN >= 1) { y = (y - m0) * (g0 * iv0) + be0; y = fmaxf(y, 0.0f); }
      if constexpr (NBN >= 2) { y = (y - m1) * (g1 * iv1) + be1; y = fmaxf(y, 0.0f); }
      if constexpr (HAS_RES)  { if (cok[t]) y += res[(size_t)r * ncols + c]; }
      if (cok[t]) out[(size_t)r * ncols + c] = y;
    }
  }
}

// ------------------------------------------------------------- pose head
// h3d[n,:] = h[n,:] @ W_out + b_out ; atomic accumulate per-graph sum + count
__global__ void pose_head_kernel(const float* __restrict__ h, const float* __restrict__ W_out,
                                 const float* __restrict__ b_out, const int* __restrict__ n2g,
                                 float* __restrict__ h3d, float* __restrict__ gsum,
                                 float* __restrict__ gcnt, int N) {
  int n = blockIdx.x * blockDim.x + threadIdx.x;
  if (n >= N) return;
  float s0 = b_out[0], s1 = b_out[1], s2 = b_out[2];
  const float* row = h + (size_t)n * HN;
#pragma unroll 4
  for (int k = 0; k < HN; ++k) {
    float xv = row[k];
    s0 += xv * W_out[k * 3 + 0];
    s1 += xv * W_out[k * 3 + 1];
    s2 += xv * W_out[k * 3 + 2];
  }
  h3d[(size_t)n * 3 + 0] = s0;
  h3d[(size_t)n * 3 + 1] = s1;
  h3d[(size_t)n * 3 + 2] = s2;
  int g = n2g[n];
  atomicAdd(&gsum[(size_t)g * 3 + 0], s0);
  atomicAdd(&gsum[(size_t)g * 3 + 1], s1);
  atomicAdd(&gsum[(size_t)g * 3 + 2], s2);
  atomicAdd(&gcnt[g], 1.0f);
}

// z0[g,h] = (gsum[g,:]/max(gcnt,1)) @ W_ci + b_ci
__global__ void gmean_ci_kernel(const float* __restrict__ gsum, const float* __restrict__ gcnt,
                                const float* __restrict__ W_ci, const float* __restrict__ b_ci,
                                float* __restrict__ z0, int G) {
  long t = (long)blockIdx.x * blockDim.x + threadIdx.x;
  if (t >= (long)G * HN) return;
  int g = (int)(t >> 8);
  int h = (int)(t & (HN - 1));
  float inv = 1.0f / fmaxf(gcnt[g], 1.0f);
  float m0 = gsum[(size_t)g * 3 + 0] * inv;
  float m1 = gsum[(size_t)g * 3 + 1] * inv;
  float m2 = gsum[(size_t)g * 3 + 2] * inv;
  z0[t] = m0 * W_ci[h] + m1 * W_ci[HN + h] + m2 * W_ci[2 * HN + h] + b_ci[h];
}

// ---------------------------------------------------------------- launcher
static inline int gemm_blocks(int rows, int ncols, int nt) {
  int gtn = (((ncols + 15) >> 4) + nt - 1) / nt;
  int tiles = (rows >> 4) * gtn;
  return (tiles + 7) / 8;   // 8 waves per 256-thread block
}

extern "C" void kernel_launch(void* const* d_in, const int* in_sizes, int n_in,
                              void* d_out, int out_size, void* d_ws, size_t ws_size,
                              hipStream_t stream) {
  const int N = 139264, E = 278528, G = 8192;
  (void)in_sizes; (void)n_in; (void)out_size; (void)ws_size;

  const float* nf    = (const float*)d_in[0];
  const int*   src   = (const int*)d_in[1];
  const int*   dst   = (const int*)d_in[2];
  const int*   n2g   = (const int*)d_in[3];
  const float* W_emb = (const float*)d_in[4];
  const float* b_emb = (const float*)d_in[5];
  const float* W_c1  = (const float*)d_in[6];
  const float* b_c1  = (const float*)d_in[7];
  const float* W_c2  = (const float*)d_in[8];
  const float* b_c2  = (const float*)d_in[9];
  const float* bnG   = (const float*)d_in[10];
  const float* bnB   = (const float*)d_in[11];
  const float* bnM   = (const float*)d_in[12];
  const float* bnV   = (const float*)d_in[13];
  const float* W_out = (const float*)d_in[14];
  const float* b_out = (const float*)d_in[15];
  const float* W_ci  = (const float*)d_in[16];
  const float* b_ci  = (const float*)d_in[17];
  const float* W_b3a = (const float*)d_in[18];
  const float* b_b3a = (const float*)d_in[19];
  const float* W_b3b = (const float*)d_in[20];
  const float* b_b3b = (const float*)d_in[21];
  const float* W_cls = (const float*)d_in[22];
  const float* b_cls = (const float*)d_in[23];

  // workspace carve-up (floats)
  float* w     = (float*)d_ws;
  float* nrm_o = w;                              // [N] out-degree -> rsqrt
  float* nrm_i = nrm_o + N;                      // [N] in-degree  -> rsqrt
  float* gsum  = nrm_i + N;                      // [G,3]
  float* gcnt  = gsum + (size_t)G * 3;           // [G]
  float* zA    = gcnt + G;                       // [G,H]
  float* zB    = zA + (size_t)G * HN;            // [G,H]
  float* buf0  = zB + (size_t)G * HN;            // [N,H] x, later h2
  float* buf1  = buf0 + (size_t)N * HN;          // [N,H] agg (conv1 & conv2)
  float* buf2  = buf1 + (size_t)N * HN;          // [N,H] h1

  float* h3d   = (float*)d_out;                  // [N,3]
  float* label = h3d + (size_t)N * 3;            // [G,60]

  const int T = 256;
  const long NH = (long)N * HN;

  // degrees -> 1/sqrt(clip(deg,1))
  zero_f32<<<2048, T, 0, stream>>>(nrm_o, 2L * N);
  zero_f32<<<256, T, 0, stream>>>(gsum, (long)G * 4);   // gsum + gcnt contiguous
  degree_kernel<<<(E + T - 1) / T, T, 0, stream>>>(src, dst, nrm_o, nrm_i, E);
  rsqrt_clip_kernel<<<(2 * N + T - 1) / T, T, 0, stream>>>(nrm_o, 2 * N);

  // embedding
  embed_kernel<<<(int)((NH + T - 1) / T), T, 0, stream>>>(nf, W_emb, b_emb, buf0, N);

  // conv block 1: scatter (with both norms folded) then fused WMMA GEMM
  zero_f32<<<8192, T, 0, stream>>>(buf1, NH);
  scatter_kernel<<<(int)(((long)E * 64 + T - 1) / T), T, 0, stream>>>(
      buf0, nrm_o, nrm_i, src, dst, buf1, E);
  wmma_gemm_fused<4, 2, true, false><<<gemm_blocks(N, HN, 4), T, 0, stream>>>(
      buf1, W_c1, b_c1, bnG, bnB, bnM, bnV, 0, 1, buf0, buf2, N, HN);

  // conv block 2
  zero_f32<<<8192, T, 0, stream>>>(buf1, NH);
  scatter_kernel<<<(int)(((long)E * 64 + T - 1) / T), T, 0, stream>>>(
      buf2, nrm_o, nrm_i, src, dst, buf1, E);
  wmma_gemm_fused<4, 2, true, false><<<gemm_blocks(N, HN, 4), T, 0, stream>>>(
      buf1, W_c2, b_c2, bnG, bnB, bnM, bnV, 2, 3, buf2, buf0, N, HN);

  // pose head + per-graph mean accumulation
  pose_head_kernel<<<(N + T - 1) / T, T, 0, stream>>>(buf0, W_out, b_out, n2g, h3d, gsum, gcnt, N);

  // classification head
  gmean_ci_kernel<<<(int)(((long)G * HN + T - 1) / T), T, 0, stream>>>(gsum, gcnt, W_ci, b_ci, zA, G);
  wmma_gemm_fused<4, 1, false, false><<<gemm_blocks(G, HN, 4), T, 0, stream>>>(
      zA, W_b3a, b_b3a, bnG, bnB, bnM, bnV, 4, 4, nullptr, zB, G, HN);
  wmma_gemm_fused<4, 1, false, false><<<gemm_blocks(G, HN, 4), T, 0, stream>>>(
      zB, W_b3b, b_b3b, bnG, bnB, bnM, bnV, 5, 5, nullptr, zA, G, HN);
  wmma_gemm_fused<4, 0, false, true><<<gemm_blocks(G, 60, 4), T, 0, stream>>>(
      zA, W_cls, b_cls, bnG, bnB, bnM, bnV, 0, 0, nullptr, label, G, 60);
}